// HopfieldMemory_62809601737235
// MI455X (gfx1250) — compile-verified
//
#include <hip/hip_runtime.h>

// ---------------------------------------------------------------------------
// HopfieldMemory target-grad kernel for MI455X (gfx1250, wave32, WMMA).
//
// B=4,H=16 (BH=64), KP=512 slots, N=4096 data, D=64.
//   S = K xk^T ; dk = softmax_k(S) ; dkx = dk / rowsum_n(dk)
//   gK = dkx@xk - K ; gV = dkx@xv - V      (alpha=0, beta=1)
//
// Row normalization commutes out of the readout GEMM, so:
//   pass 1: online column stats m[n], Z[n]            (S computed once)
//   pass 2: p=exp(S-m)/Z ; acc += p@[xk,xv]; r[k]+=p  (S recomputed once)
//           out = acc / r[k] - {K,V}
// Memory-bound (~164 MB @ 23.3 TB/s); exact f32 WMMA (16x16x4) is free.
// ---------------------------------------------------------------------------

typedef __attribute__((ext_vector_type(2))) float v2f;
typedef __attribute__((ext_vector_type(8))) float v8f;

#define BH   64
#define KP   512
#define NTOT 4096
#define DD   64
#define KT   (KP / 16)    // 32 k-tiles
#define NT   (NTOT / 16)  // 256 n-tiles
#define WAVES 8           // 256 threads / block
#define LDSP  18          // padded row stride (floats) for p-tile staging

static __device__ __forceinline__ v8f wmma_f32(v2f a, v2f b, v8f c) {
  // D = A(16x4,f32) * B(4x16,f32) + C(16x16,f32)
  return __builtin_amdgcn_wmma_f32_16x16x4_f32(false, a, false, b,
                                               (short)0, c, false, false);
}

// ---------------------------------------------------------------------------
// Pass 1: per-column (n) softmax stats over all k=512, online (flash-style).
// One wave per 16-wide n strip of one (b,h). stats[bh*N + n] = (m, Z).
// ---------------------------------------------------------------------------
__global__ __launch_bounds__(256) void hm_stats(const float* __restrict__ K,
                                                const float* __restrict__ xk,
                                                float2* __restrict__ stats) {
  const int wave = threadIdx.x >> 5;
  const int lane = threadIdx.x & 31;
  const int lh   = lane >> 4;      // lane half (0/1)
  const int ln   = lane & 15;      // position within half
  const int gw   = blockIdx.x * WAVES + wave;
  const int bh   = gw / NT;
  const int n0   = (gw % NT) * 16;

  // B operand: xk^T strip (64 x 16), fixed for this wave.
  // chunk c: lane holds B[d=4c+2*lh .. +1][n=ln] = xk[n][d..d+1] -> float2
  v2f b[16];
  {
    const float* xb = xk + ((size_t)bh * NTOT + n0 + ln) * DD + 2 * lh;
#pragma unroll
    for (int c = 0; c < 16; ++c) b[c] = *(const v2f*)(xb + 4 * c);
  }

  float m = -INFINITY, Z = 0.f;
  const float* Kb = K + (size_t)bh * KP * DD + 2 * lh;

  for (int kt = 0; kt < KT; ++kt) {
    // A operand: K tile (16 x 64). chunk c: lane holds A[M=ln][d=4c+2*lh..+1]
    const float* Ka = Kb + (size_t)(kt * 16 + ln) * DD;
    v2f a[16];
#pragma unroll
    for (int c = 0; c < 16; ++c) a[c] = *(const v2f*)(Ka + 4 * c);

    v8f s = {};
#pragma unroll
    for (int c = 0; c < 16; ++c) s = wmma_f32(a[c], b[c], s);

    // column max over the 16 rows of this tile (8 in each lane half)
    float tmax = s[0];
#pragma unroll
    for (int j = 1; j < 8; ++j) tmax = fmaxf(tmax, s[j]);
    tmax = fmaxf(tmax, __shfl_xor(tmax, 16, 32));

    const float mn = fmaxf(m, tmax);
    float esum = 0.f;
#pragma unroll
    for (int j = 0; j < 8; ++j) esum += __expf(s[j] - mn);
    esum += __shfl_xor(esum, 16, 32);

    Z = Z * __expf(m - mn) + esum;
    m = mn;
  }
  if (lh == 0) stats[(size_t)bh * NTOT + n0 + ln] = make_float2(m, Z);
}

// ---------------------------------------------------------------------------
// Pass 2: one wave per 16-row k-tile of one (b,h); sweeps all 256 n-tiles.
// Accumulates acc_K/acc_V (16x64 each) + row sums in registers, then scales,
// subtracts K/V and stores gK/gV.
// ---------------------------------------------------------------------------
__global__ __launch_bounds__(256) void hm_update(const float* __restrict__ K,
                                                 const float* __restrict__ V,
                                                 const float* __restrict__ xk,
                                                 const float* __restrict__ xv,
                                                 const float2* __restrict__ stats,
                                                 float* __restrict__ out) {
  __shared__ float pstage[WAVES][16 * LDSP];  // per-wave p-tile transpose buf

  const int wave = threadIdx.x >> 5;
  const int lane = threadIdx.x & 31;
  const int lh   = lane >> 4;
  const int ln   = lane & 15;
  const int gw   = blockIdx.x * WAVES + wave;
  const int bh   = gw / KT;
  const int k0   = (gw % KT) * 16;

  float* P = pstage[wave];

  // A operand for GEMM1: this wave's K tile (16 x 64), resident in VGPRs.
  v2f a[16];
  {
    const float* Ka = K + ((size_t)bh * KP + k0 + ln) * DD + 2 * lh;
#pragma unroll
    for (int c = 0; c < 16; ++c) a[c] = *(const v2f*)(Ka + 4 * c);
  }

  v8f accK[4] = {};          // 4 d-tiles of acc_K (16x16 each)
  v8f accV[4] = {};
  float racc[8] = {};        // per-lane partial row sums (C layout rows)

  const float* xkH = xk + (size_t)bh * NTOT * DD;
  const float* xvH = xv + (size_t)bh * NTOT * DD;
  const float2* st = stats + (size_t)bh * NTOT;

  for (int nt = 0; nt < NT; ++nt) {
    const int n0 = nt * 16;

    // GEMM1: S tile = K_tile * xk^T strip
    v8f s = {};
    {
      const float* xb = xkH + (size_t)(n0 + ln) * DD + 2 * lh;
#pragma unroll
      for (int c = 0; c < 16; ++c) {
        v2f b = *(const v2f*)(xb + 4 * c);
        s = wmma_f32(a[c], b, s);
      }
    }

    // p = exp(S - m[n]) / Z[n]   (beta = 1)
    const float2 mz = st[n0 + ln];
    const float rz = 1.0f / mz.y;
    float p[8];
#pragma unroll
    for (int j = 0; j < 8; ++j) {
      p[j] = __expf(s[j] - mz.x) * rz;
      racc[j] += p[j];
    }

    // C layout -> A layout transpose through wave-private LDS tile
#pragma unroll
    for (int j = 0; j < 8; ++j) P[(j + 8 * lh) * LDSP + ln] = p[j];
    asm volatile("s_wait_dscnt 0" ::: "memory");  // wave-local RAW on LDS

    v2f pa[4];  // A operand chunks over n: A[M=ln][n=4c+2*lh .. +1]
#pragma unroll
    for (int c = 0; c < 4; ++c)
      pa[c] = *(const v2f*)&P[ln * LDSP + 4 * c + 2 * lh];

    // GEMM2: acc += p_tile * xk / xv   (K-dim = 16 -> 4 chunks of 4)
    const float* xkB = xkH + (size_t)n0 * DD;
    const float* xvB = xvH + (size_t)n0 * DD;
#pragma unroll
    for (int dt = 0; dt < 4; ++dt) {
#pragma unroll
      for (int c = 0; c < 4; ++c) {
        const int row = 4 * c + 2 * lh;          // n within tile
        const int col = dt * 16 + ln;            // d
        v2f bk, bv;
        bk.x = xkB[(size_t)row * DD + col];
        bk.y = xkB[(size_t)(row + 1) * DD + col];
        bv.x = xvB[(size_t)row * DD + col];
        bv.y = xvB[(size_t)(row + 1) * DD + col];
        accK[dt] = wmma_f32(pa[c], bk, accK[dt]);
        accV[dt] = wmma_f32(pa[c], bv, accV[dt]);
      }
    }
  }

  // Reduce row partial sums within each 16-lane half; result lands exactly in
  // the C-layout row mapping (lanes 0-15 -> rows j, lanes 16-31 -> rows j+8).
#pragma unroll
  for (int j = 0; j < 8; ++j) {
    float r = racc[j];
    r += __shfl_xor(r, 1, 32);
    r += __shfl_xor(r, 2, 32);
    r += __shfl_xor(r, 4, 32);
    r += __shfl_xor(r, 8, 32);
    racc[j] = 1.0f / r;
  }

  // gK = acc_K / r - K ; gV = acc_V / r - V   (alpha=0 -> factor 1)
  const size_t gvOfs = (size_t)BH * KP * DD;
#pragma unroll
  for (int dt = 0; dt < 4; ++dt) {
#pragma unroll
    for (int j = 0; j < 8; ++j) {
      const int row = j + 8 * lh;
      const size_t idx =
          ((size_t)bh * KP + k0 + row) * DD + dt * 16 + ln;
      out[idx]         = accK[dt][j] * racc[j] - K[idx];
      out[gvOfs + idx] = accV[dt][j] * racc[j] - V[idx];
    }
  }
}

// ---------------------------------------------------------------------------
extern "C" void kernel_launch(void* const* d_in, const int* in_sizes, int n_in,
                              void* d_out, int out_size, void* d_ws, size_t ws_size,
                              hipStream_t stream) {
  (void)in_sizes; (void)n_in; (void)out_size; (void)ws_size;
  const float* K  = (const float*)d_in[0];
  const float* V  = (const float*)d_in[1];
  const float* xk = (const float*)d_in[2];
  const float* xv = (const float*)d_in[3];
  float* out      = (float*)d_out;
  float2* stats   = (float2*)d_ws;  // 64*4096 float2 = 2 MB

  // Pass 1: 64 bh * 256 n-strips, 8 waves/block -> 2048 blocks
  hm_stats<<<BH * NT / WAVES, 32 * WAVES, 0, stream>>>(K, xk, stats);
  // Pass 2: 64 bh * 32 k-tiles, 8 waves/block -> 256 blocks
  hm_update<<<BH * KT / WAVES, 32 * WAVES, 0, stream>>>(K, V, xk, xv, stats, out);
}